// IterativeStrangeLoop_43224550867178
// MI455X (gfx1250) — compile-verified
//
#include <hip/hip_runtime.h>

typedef __attribute__((ext_vector_type(2))) float v2f;
typedef __attribute__((ext_vector_type(8))) float v8f;

// D(16x16,f32) = A(16x4,f32) x B(4x16,f32) + C   -- full fp32 CDNA5 WMMA
static __device__ __forceinline__ v8f wmma4(v2f a, v2f b, v8f c) {
  return __builtin_amdgcn_wmma_f32_16x16x4_f32(false, a, false, b, (short)0, c,
                                               false, false);
}

static __device__ __forceinline__ float xor16(float x) {
  return __shfl_xor(x, 16, 32);  // swap lane halves within wave32
}

// gfx1250 hardware transcendental tanh (TRANS pipe, co-executes with WMMA)
static __device__ __forceinline__ float htanh(float x) {
  float y;
  asm("v_tanh_f32 %0, %1" : "=v"(y) : "v"(x));
  return y;
}

static __device__ __forceinline__ float gelu_tanh(float x) {
  float u = 0.7978845608028654f * (x + 0.044715f * x * x * x);
  return 0.5f * x * (1.0f + htanh(u));
}

// Branchless guarded load: unconditional load from clamped index + cndmask.
static __device__ __forceinline__ float ldg(const float* __restrict__ p,
                                            int idx, bool pred) {
  float v = p[pred ? idx : 0];
  return pred ? v : 0.0f;
}

__global__ __launch_bounds__(256, 1)
void strange_loop_fp32_wmma(const float* __restrict__ s7,
                            const float* __restrict__ fc0_w,   // [14,64]
                            const float* __restrict__ fc0_b,   // [64]
                            const float* __restrict__ ln_s,    // [64]
                            const float* __restrict__ ln_b,    // [64]
                            const float* __restrict__ out_w,   // [64,7]
                            const float* __restrict__ out_b,   // [7]
                            float* __restrict__ mu_out,        // [B,7]
                            float* __restrict__ conv_out,      // [B]
                            int nrows) {
  const int lane = threadIdx.x & 31;
  const int wave = threadIdx.x >> 5;
  const int lm   = lane & 15;          // N index (batch row in tile) / A-matrix M
  const bool hi  = lane >= 16;
  const int kadd = hi ? 2 : 0;         // K = 4*b + v + 2*half  (A & B f32 layout)
  int row = blockIdx.x * 128 + wave * 16 + lm;
  const bool valid = row < nrows;
  if (row >= nrows) row = nrows - 1;   // clamp loads, keep EXEC uniform for WMMA

  // ---------- iteration-invariant A operands (weights, transposed) ----------
  // Stage-1 A: W_mu^T tiles  (M = hidden = 16t+lm, K = mu index)
  v2f a1[4][2];
#pragma unroll
  for (int t = 0; t < 4; ++t)
#pragma unroll
    for (int b = 0; b < 2; ++b) {
      int k0 = 4 * b + kadd;
      a1[t][b].x = ldg(fc0_w, (k0    ) * 64 + 16 * t + lm, k0     < 7);
      a1[t][b].y = ldg(fc0_w, (k0 + 1) * 64 + 16 * t + lm, k0 + 1 < 7);
    }
  // Stage-2 A: W1^T  (M = outdim = lm (pad 16), K = hidden)
  v2f a2[16];
#pragma unroll
  for (int b = 0; b < 16; ++b) {
    int k0 = 4 * b + kadd;
    a2[b].x = ldg(out_w, (k0    ) * 7 + lm, lm < 7);
    a2[b].y = ldg(out_w, (k0 + 1) * 7 + lm, lm < 7);
  }
  // LayerNorm affine params per D1 element (M = hidden = 16t + r + 8*half)
  v8f lnS[4], lnB[4];
#pragma unroll
  for (int t = 0; t < 4; ++t)
#pragma unroll
    for (int r = 0; r < 8; ++r) {
      int h = 16 * t + r + (hi ? 8 : 0);
      lnS[t][r] = ln_s[h];
      lnB[t][r] = ln_b[h];
    }
  // Stage-2 C template: out bias at (M=outdim)
  float ob[8];
#pragma unroll
  for (int r = 0; r < 8; ++r) ob[r] = ldg(out_b, r, !hi && r < 7);

  // ---------- base = W_s7^T @ s7^T + fc0_b  (iteration-invariant D tiles) ----
  v2f bs7[2];
#pragma unroll
  for (int b = 0; b < 2; ++b) {
    int k0 = 4 * b + kadd;
    long roff = (long)row * 7;
    bs7[b].x = ldg(s7, (int)roff + k0,     k0     < 7);
    bs7[b].y = ldg(s7, (int)roff + k0 + 1, k0 + 1 < 7);
  }
  v8f base[4];
#pragma unroll
  for (int t = 0; t < 4; ++t) {
    v8f c;
#pragma unroll
    for (int r = 0; r < 8; ++r) c[r] = fc0_b[16 * t + r + (hi ? 8 : 0)];
#pragma unroll
    for (int b = 0; b < 2; ++b) {
      int k0 = 4 * b + kadd;
      v2f a;
      a.x = ldg(fc0_w, (7 + k0    ) * 64 + 16 * t + lm, k0     < 7);
      a.y = ldg(fc0_w, (7 + k0 + 1) * 64 + 16 * t + lm, k0 + 1 < 7);
      c = wmma4(a, bs7[b], c);
    }
    base[t] = c;
  }

  // ---------- fixed-point loop ----------
  float mu[7];
#pragma unroll
  for (int j = 0; j < 7; ++j) mu[j] = 0.3779644730092272f;  // 1/sqrt(7)
  float conv = 0.0f;

#pragma unroll 1
  for (int it = 0; it < 10; ++it) {
    // B1 = mu^T (K = mu idx, N = row); mu replicated in both lane halves
    v2f b1[2];
    b1[0].x = hi ? mu[2] : mu[0];
    b1[0].y = hi ? mu[3] : mu[1];
    b1[1].x = hi ? mu[6] : mu[4];
    b1[1].y = hi ? 0.0f  : mu[5];

    // D1[t] = base[t] + W_mu^T @ mu^T   (h layout: M=hidden, N=row)
    v8f h[4];
#pragma unroll
    for (int t = 0; t < 4; ++t) {
      v8f c = base[t];
      c = wmma4(a1[t][0], b1[0], c);
      c = wmma4(a1[t][1], b1[1], c);
      h[t] = c;
    }

    // LayerNorm over hidden (per batch row = lane pair): 1 shuffle per moment
    float s = 0.0f, sq = 0.0f;
#pragma unroll
    for (int t = 0; t < 4; ++t)
#pragma unroll
      for (int r = 0; r < 8; ++r) {
        float x = h[t][r];
        s += x;
        sq += x * x;
      }
    s += xor16(s);
    sq += xor16(sq);
    float mean = s * (1.0f / 64.0f);
    float var  = sq * (1.0f / 64.0f) - mean * mean;
    float rstd = rsqrtf(var + 1e-6f);
#pragma unroll
    for (int t = 0; t < 4; ++t)
#pragma unroll
      for (int r = 0; r < 8; ++r) {
        float x = (h[t][r] - mean) * rstd * lnS[t][r] + lnB[t][r];
        h[t][r] = gelu_tanh(x);
      }

    // Stage 2: D2 = W1^T @ g^T + out_b. Build B2 (K=hidden, N=row) from the
    // D-layout g via cross-half shuffles (register-only transpose).
    v8f d2;
#pragma unroll
    for (int r = 0; r < 8; ++r) d2[r] = ob[r];
#pragma unroll
    for (int b = 0; b < 16; ++b) {
      v2f bb;
#pragma unroll
      for (int v = 0; v < 2; ++v) {
        int j0 = 4 * b + v;          // hidden needed by low half (+2 for high)
        int t = j0 >> 4;
        int q = j0 & 15;
        float val;
        if (q < 8) {                 // both targets live in low lanes
          float oth = xor16(h[t][q + 2]);
          val = hi ? oth : h[t][q];
        } else {                     // both targets live in high lanes
          int rA = q - 8;
          float oth = xor16(h[t][rA]);
          val = hi ? h[t][rA + 2] : oth;
        }
        if (v == 0) bb.x = val; else bb.y = val;
      }
      d2 = wmma4(a2[b], bb, d2);
    }

    // upd / normalize / damp / convergence: fully lane-local (low lanes own it)
    float n2 = 0.0f;
    float u[7];
#pragma unroll
    for (int j = 0; j < 7; ++j) {
      u[j] = d2[j];
      n2 += u[j] * u[j];
    }
    float inv = 1.0f / (sqrtf(n2) + 1e-8f);
    float dsum = 0.0f;
    float mun[7];
#pragma unroll
    for (int j = 0; j < 7; ++j) {
      float un = u[j] * inv;
      float mn = 0.5f * (mu[j] + un);
      float dd = mn - mu[j];
      dsum += dd * dd;
      mun[j] = mn;
    }
    conv = (sqrtf(dsum) < 1e-4f) ? 1.0f : conv;
    // broadcast updated mu from low half to both halves for next iteration
#pragma unroll
    for (int j = 0; j < 7; ++j) {
      float t = xor16(mun[j]);
      mu[j] = hi ? t : mun[j];
    }
  }

  // ---------- store (low lanes hold the real rows) ----------
  if (!hi && valid) {
#pragma unroll
    for (int j = 0; j < 7; ++j) mu_out[(size_t)row * 7 + j] = mu[j];
    conv_out[row] = conv;
  }
}

extern "C" void kernel_launch(void* const* d_in, const int* in_sizes, int n_in,
                              void* d_out, int out_size, void* d_ws,
                              size_t ws_size, hipStream_t stream) {
  (void)n_in; (void)out_size; (void)d_ws; (void)ws_size;
  const float* s7    = (const float*)d_in[0];
  const float* fc0_w = (const float*)d_in[1];
  const float* fc0_b = (const float*)d_in[2];
  const float* ln_s  = (const float*)d_in[3];
  const float* ln_b  = (const float*)d_in[4];
  const float* out_w = (const float*)d_in[5];
  const float* out_b = (const float*)d_in[6];
  const int B = in_sizes[0] / 7;
  float* mu_out   = (float*)d_out;
  float* conv_out = mu_out + (size_t)B * 7;
  const int blocks = (B + 127) / 128;  // 8 waves/block, 16 rows/wave
  hipLaunchKernelGGL(strange_loop_fp32_wmma, dim3(blocks), dim3(256), 0, stream,
                     s7, fc0_w, fc0_b, ln_s, ln_b, out_w, out_b, mu_out,
                     conv_out, B);
}